// energy_dc_bc_loss_61375082659858
// MI455X (gfx1250) — compile-verified
//
#include <hip/hip_runtime.h>

// Problem constants (reference: img (8,3,256,256) f32, y_pred (8,1,256,256) f32)
#define BB    8
#define HH    256
#define WW    256
#define HW    65536       // HH*WW
#define BHW   524288      // BB*HW
#define RAD   7           // 15x15 window radius
#define NPIX  65          // int(0.001*HW)
#define NPOOL 520         // BB*NPIX (reference pools indices across batches for A2)
#define HV    254
#define NV    64516       // HV*HV
#define BNV   516128      // BB*NV
#define NRED  512         // partial blocks for global reductions

typedef __attribute__((ext_vector_type(2))) float v2f;
typedef __attribute__((ext_vector_type(8))) float v8f;

// ---------------------------------------------------------------------------
// Wave-level sum via V_WMMA_F32_16X16X4_F32 (CDNA5 matrix path, f32-exact).
// A (16x4): lanes 0-15 hold K=0 in VGPR0 (our x), lanes 16-31 hold K=2 (x).
// K=1,3 are zero. B = ones(4x16)  =>  D[m,n] = x_m + x_{m+16}.
// Total wave sum = sum over M=0..15 = sum_r ( D@lane0 (M=r) + D@lane16 (M=8+r) ).
// ---------------------------------------------------------------------------
__device__ __forceinline__ float wave_sum_wmma(float x) {
  v2f a; a.x = x;    a.y = 0.0f;
  v2f b; b.x = 1.0f; b.y = 1.0f;
  v8f c = {};
  c = __builtin_amdgcn_wmma_f32_16x16x4_f32(false, a, false, b, (short)0, c,
                                            false, false);
  float s = 0.0f;
#pragma unroll
  for (int i = 0; i < 8; ++i)
    s += __shfl(c[i], 0, 32) + __shfl(c[i], 16, 32);
  return s;
}

// Block (256 threads = 8 waves) sum; sh must hold >= 8 floats. Result on all threads.
__device__ __forceinline__ float block_sum256(float x, float* sh) {
  float w = wave_sum_wmma(x);
  if ((threadIdx.x & 31) == 0) sh[threadIdx.x >> 5] = w;
  __syncthreads();
  float tot = sh[0] + sh[1] + sh[2] + sh[3] + sh[4] + sh[5] + sh[6] + sh[7];
  __syncthreads();  // allow shared-mem reuse by subsequent calls
  return tot;
}

// ---------------------------------------------------------------------------
// Stage A: per-pixel channel min & max of img
// ---------------------------------------------------------------------------
__global__ void k_chan_minmax(const float* __restrict__ img,
                              float* __restrict__ cmin, float* __restrict__ cmax) {
  int i = blockIdx.x * blockDim.x + threadIdx.x;
  if (i >= BHW) return;
  int b = i / HW, p = i - b * HW;
  const float* s = img + (size_t)b * 3 * HW + p;
  float a0 = s[0], a1 = s[HW], a2 = s[2 * HW];
  cmin[i] = fminf(a0, fminf(a1, a2));
  cmax[i] = fmaxf(a0, fmaxf(a1, a2));
}

// 15-tap horizontal window pass (OOB skipped, matching -inf/+inf padding)
template <bool ISMIN>
__global__ void k_row_win(const float* __restrict__ in, float* __restrict__ out) {
  int i = blockIdx.x * blockDim.x + threadIdx.x;
  if (i >= BHW) return;
  int x = i % WW;
  int x0 = (x - RAD < 0) ? 0 : x - RAD;
  int x1 = (x + RAD > WW - 1) ? WW - 1 : x + RAD;
  const float* row = in + (i - x);
  float v = row[x0];
  for (int xx = x0 + 1; xx <= x1; ++xx)
    v = ISMIN ? fminf(v, row[xx]) : fmaxf(v, row[xx]);
  out[i] = v;
}

// 15-tap vertical window pass
template <bool ISMIN>
__global__ void k_col_win(const float* __restrict__ in, float* __restrict__ out) {
  int i = blockIdx.x * blockDim.x + threadIdx.x;
  if (i >= BHW) return;
  int b = i / HW, r = i - b * HW;
  int y = r / WW, x = r - y * WW;
  int y0 = (y - RAD < 0) ? 0 : y - RAD;
  int y1 = (y + RAD > HH - 1) ? HH - 1 : y + RAD;
  const float* base = in + (size_t)b * HW + x;
  float v = base[y0 * WW];
  for (int yy = y0 + 1; yy <= y1; ++yy)
    v = ISMIN ? fminf(v, base[yy * WW]) : fmaxf(v, base[yy * WW]);
  out[i] = v;
}

// ---------------------------------------------------------------------------
// Top-65 selection (one block per batch). findMax=1: top-65 of dark -> A1.
// findMax=0: bottom-65 of bright -> global pixel indices (for pooled A2).
// vals is workspace-resident and mutated (masked with +/-FLT_MAX).
// ---------------------------------------------------------------------------
__global__ void k_select_extreme(float* __restrict__ vals,
                                 const float* __restrict__ img, int findMax,
                                 float* __restrict__ A1, int* __restrict__ idxout) {
  __shared__ float sv[256];
  __shared__ int   si[256];
  __shared__ int   chosen[NPIX];
  int b = blockIdx.x, tid = threadIdx.x;
  float* d = vals + (size_t)b * HW;

  for (int j = 0; j < NPIX; ++j) {
    float best = findMax ? -3.4e38f : 3.4e38f;
    int bi = HW;
    for (int p = tid; p < HW; p += 256) {
      float v = d[p];
      bool better = findMax ? (v > best) : (v < best);  // strict: earliest index wins
      if (better) { best = v; bi = p; }
    }
    sv[tid] = best; si[tid] = bi;
    __syncthreads();
    for (int s = 128; s > 0; s >>= 1) {
      if (tid < s) {
        float v2 = sv[tid + s]; int i2 = si[tid + s];
        bool better = findMax ? (v2 > sv[tid] || (v2 == sv[tid] && i2 < si[tid]))
                              : (v2 < sv[tid] || (v2 == sv[tid] && i2 < si[tid]));
        if (better) { sv[tid] = v2; si[tid] = i2; }
      }
      __syncthreads();
    }
    if (tid == 0) {
      int p = si[0];
      chosen[j] = p;
      d[p] = findMax ? -3.4e38f : 3.4e38f;  // mask
    }
    __threadfence_block();
    __syncthreads();
  }

  if (tid == 0) {
    if (findMax) {
      // A1 = I at the pixel (among top-65 dark) with the largest max-channel value.
      const float* ib = img + (size_t)b * 3 * HW;
      float bm = -3.4e38f; int bp = chosen[0];
      for (int j = 0; j < NPIX; ++j) {
        int p = chosen[j];
        float mx = fmaxf(ib[p], fmaxf(ib[p + HW], ib[p + 2 * HW]));
        if (mx > bm) { bm = mx; bp = p; }  // strict: first max (top_k order)
      }
      A1[b * 3 + 0] = ib[bp];
      A1[b * 3 + 1] = ib[bp + HW];
      A1[b * 3 + 2] = ib[bp + 2 * HW];
    } else {
      for (int j = 0; j < NPIX; ++j) idxout[b * NPIX + j] = chosen[j];
    }
  }
}

// A2[b,c] = mean over ALL batches' 65 indices (520 pooled) of img[b,c,idx]
__global__ void k_A2(const float* __restrict__ img, const int* __restrict__ idx,
                     float* __restrict__ A2) {
  int blk = blockIdx.x;          // BB*3 blocks
  int b = blk / 3, c = blk % 3;
  const float* ib = img + (size_t)b * 3 * HW + (size_t)c * HW;
  __shared__ float sh[64];
  float acc = 0.0f;
  for (int j = threadIdx.x; j < NPOOL; j += 64) acc += ib[idx[j]];
  sh[threadIdx.x] = acc;
  __syncthreads();
  for (int s = 32; s > 0; s >>= 1) {
    if (threadIdx.x < s) sh[threadIdx.x] += sh[threadIdx.x + s];
    __syncthreads();
  }
  if (threadIdx.x == 0) A2[b * 3 + c] = sh[0] / (float)NPOOL;
}

// per-pixel: n1 = min_c(img/A1_c), n2 = max_c((1-img)/(1-A2_c))
__global__ void k_chan_norm(const float* __restrict__ img,
                            const float* __restrict__ A1, const float* __restrict__ A2,
                            float* __restrict__ n1, float* __restrict__ n2) {
  int i = blockIdx.x * blockDim.x + threadIdx.x;
  if (i >= BHW) return;
  int b = i / HW, p = i - b * HW;
  const float* s = img + (size_t)b * 3 * HW + p;
  float v1 = 3.4e38f, v2 = -3.4e38f;
#pragma unroll
  for (int c = 0; c < 3; ++c) {
    float x = s[c * HW];
    v1 = fminf(v1, x / A1[b * 3 + c]);
    v2 = fmaxf(v2, (1.0f - x) / (1.0f - A2[b * 3 + c]));
  }
  n1[i] = v1;
  n2[i] = v2;
}

// Fused: vertical window pass + data term  sum (y - (1 + coef*win))^2
// coef = -OMEGA for t_slide1, coef = -1 for t_slide2.
template <bool ISMIN>
__global__ void k_col_win_data(const float* __restrict__ in,
                               const float* __restrict__ ypred, float coef,
                               float* __restrict__ partials) {
  float acc = 0.0f;
  int stride = gridDim.x * blockDim.x;
  for (int i = blockIdx.x * blockDim.x + threadIdx.x; i < BHW; i += stride) {
    int b = i / HW, r = i - b * HW;
    int y = r / WW, x = r - y * WW;
    int y0 = (y - RAD < 0) ? 0 : y - RAD;
    int y1 = (y + RAD > HH - 1) ? HH - 1 : y + RAD;
    const float* base = in + (size_t)b * HW + x;
    float v = base[y0 * WW];
    for (int yy = y0 + 1; yy <= y1; ++yy)
      v = ISMIN ? fminf(v, base[yy * WW]) : fmaxf(v, base[yy * WW]);
    float t = 1.0f + coef * v;
    float d = ypred[i] - t;
    acc += d * d;
  }
  __shared__ float sh[8];
  float tot = block_sum256(acc, sh);
  if (threadIdx.x == 0) partials[blockIdx.x] = tot;
}

// Smoothness: nd = 18*S2 - 2*S1^2 over 3x3 patches of y_pred (valid region).
// (w_sum == 9 algebraically -- applied as a constant in the final kernel.)
__global__ void k_smooth(const float* __restrict__ y, float* __restrict__ partials) {
  float acc = 0.0f;
  int stride = gridDim.x * blockDim.x;
  for (int i = blockIdx.x * blockDim.x + threadIdx.x; i < BNV; i += stride) {
    int b = i / NV, r = i - b * NV;
    int yy = r / HV, xx = r - yy * HV;
    const float* p0 = y + (size_t)b * HW + yy * WW + xx;
    float S1 = 0.0f, S2 = 0.0f;
#pragma unroll
    for (int di = 0; di < 3; ++di)
#pragma unroll
      for (int dj = 0; dj < 3; ++dj) {
        float v = p0[di * WW + dj];
        S1 += v;
        S2 += v * v;
      }
    acc += 18.0f * S2 - 2.0f * S1 * S1;
  }
  __shared__ float sh[8];
  float tot = block_sum256(acc, sh);
  if (threadIdx.x == 0) partials[blockIdx.x] = tot;
}

// Final: reduce the three partial arrays (NRED each) and combine the loss.
__global__ void k_final(const float* __restrict__ pd1, const float* __restrict__ pd2,
                        const float* __restrict__ psm, float* __restrict__ out) {
  __shared__ float sh[8];
  int tid = threadIdx.x;
  float d1 = block_sum256(pd1[tid] + pd1[tid + 256], sh);
  float d2 = block_sum256(pd2[tid] + pd2[tid + 256], sh);
  float sm = block_sum256(psm[tid] + psm[tid + 256], sh);
  if (tid == 0) {
    float smoothness = 9.0f * sm;          // w_sum == 9 (sum of centered vals == 0)
    float invN = 1.0f / (float)NV;
    float loss1 = (1.0f * smoothness + 1e-4f * d1) * invN;   // LAMBDA1*0.5 = 1.0
    float loss2 = (0.01f * smoothness + d2) * invN;
    out[0] = loss1 + 1.0f * loss2;                            // K = 1
  }
}

// ---------------------------------------------------------------------------
extern "C" void kernel_launch(void* const* d_in, const int* in_sizes, int n_in,
                              void* d_out, int out_size, void* d_ws, size_t ws_size,
                              hipStream_t stream) {
  const float* img   = (const float*)d_in[0];  // (8,3,256,256)
  const float* ypred = (const float*)d_in[1];  // (8,1,256,256)

  float* ws    = (float*)d_ws;                 // ~10.5 MB used
  float* cmin  = ws;                           // BHW (later reused as n1)
  float* cmax  = ws + (size_t)BHW;             // BHW (later reused as n2)
  float* tmp   = ws + 2 * (size_t)BHW;         // BHW
  float* dark  = ws + 3 * (size_t)BHW;         // BHW (mutated by selection)
  float* brgt  = ws + 4 * (size_t)BHW;         // BHW (mutated by selection)
  float* A1    = ws + 5 * (size_t)BHW;         // 24
  float* A2    = A1 + 32;                      // 24
  float* pd1   = A2 + 32;                      // NRED
  float* pd2   = pd1 + NRED;                   // NRED
  float* psm   = pd2 + NRED;                   // NRED
  int*   idx2  = (int*)(psm + NRED);           // NPOOL ints

  dim3 blk(256);
  const int gbhw = BHW / 256;  // 2048, exact

  // Stage A: dark & bright channels (separable 15x15 + channel reduce)
  k_chan_minmax<<<gbhw, blk, 0, stream>>>(img, cmin, cmax);
  k_row_win<true ><<<gbhw, blk, 0, stream>>>(cmin, tmp);
  k_col_win<true ><<<gbhw, blk, 0, stream>>>(tmp, dark);
  k_row_win<false><<<gbhw, blk, 0, stream>>>(cmax, tmp);
  k_col_win<false><<<gbhw, blk, 0, stream>>>(tmp, brgt);

  // Stage B: atmospheric lights
  k_select_extreme<<<BB, blk, 0, stream>>>(dark, img, 1, A1, idx2);
  k_select_extreme<<<BB, blk, 0, stream>>>(brgt, img, 0, A1, idx2);
  k_A2<<<BB * 3, 64, 0, stream>>>(img, idx2, A2);

  // Stage C: t_slide1/t_slide2 window passes fused with data terms
  k_chan_norm<<<gbhw, blk, 0, stream>>>(img, A1, A2, cmin, cmax);
  k_row_win<true ><<<gbhw, blk, 0, stream>>>(cmin, tmp);
  k_col_win_data<true ><<<NRED, blk, 0, stream>>>(tmp, ypred, -0.95f, pd1);
  k_row_win<false><<<gbhw, blk, 0, stream>>>(cmax, tmp);
  k_col_win_data<false><<<NRED, blk, 0, stream>>>(tmp, ypred, -1.0f, pd2);

  // Stage D: smoothness + final combine
  k_smooth<<<NRED, blk, 0, stream>>>(ypred, psm);
  k_final<<<1, blk, 0, stream>>>(pd1, pd2, psm, (float*)d_out);
}